// RodriguesLayer_66340064854455
// MI455X (gfx1250) — compile-verified
//
#include <hip/hip_runtime.h>
#include <hip/hip_bf16.h>

// Problem constants (from reference)
#define Bn 128
#define Ln 32
#define Jn 31
#define Cn 256
#define CJn 128
#define FEAT 4096   // C*16
#define APAD 20     // row stride (floats) for LDS tiles: 16 + 4 pad
                    //  - keeps 16B alignment for b128 stores (20*4=80B)
                    //  - m*20 mod 64 distinct across 16 consecutive rows (bank spread)

typedef __attribute__((ext_vector_type(2))) float v2f;
typedef __attribute__((ext_vector_type(8))) float v8f;

// ---- optional CDNA5 async global->LDS path ---------------------------------
#if defined(__has_builtin)
#if __has_builtin(__builtin_amdgcn_global_load_async_to_lds_b128)
#define HAVE_ASYNC_LDS 1
#else
#define HAVE_ASYNC_LDS 0
#endif
#else
#define HAVE_ASYNC_LDS 0
#endif

#if HAVE_ASYNC_LDS
typedef int v4i_ __attribute__((vector_size(4 * sizeof(int))));
typedef __attribute__((address_space(1))) v4i_* g_v4i_p;   // global
typedef __attribute__((address_space(3))) v4i_* l_v4i_p;   // LDS

__device__ __forceinline__ void async_copy16(const float* gsrc, float* ldst) {
    __builtin_amdgcn_global_load_async_to_lds_b128(
        (g_v4i_p)(gsrc), (l_v4i_p)(ldst), 0, 0);
}
#endif

__device__ __forceinline__ void wait_async0() {
#if defined(__has_builtin) && __has_builtin(__builtin_amdgcn_s_wait_asynccnt)
    __builtin_amdgcn_s_wait_asynccnt(0);
#else
    asm volatile("s_wait_asynccnt 0x0" ::: "memory");
#endif
}

// ---------------------------------------------------------------------------
// Kernel 1: Kmat[b,j,16] = joint_feats[b,j,:] @ Kw[j,k,:] + Kb[j,k]
//           Kmat2 = Kmat @ Kmat  (4x4), both to workspace.
// ---------------------------------------------------------------------------
__global__ __launch_bounds__(256)
void kmat_kernel(const float* __restrict__ joint, const float* __restrict__ Kw,
                 const float* __restrict__ Kb, float* __restrict__ kmat,
                 float* __restrict__ kmat2)
{
    __shared__ float sK[256];
    const int g = blockIdx.x * 256 + threadIdx.x;      // < 63488
    const int p = g >> 4;                              // pair (b,j)
    const int e = g & 15;                              // output element
    const int b = p / Jn;
    const int j = p - b * Jn;

    const float* jf = joint + ((size_t)b * Jn + j) * CJn;
    const float* kw = Kw + ((size_t)j * 16 + e) * CJn;
    float acc = Kb[j * 16 + e];
#pragma unroll 8
    for (int t = 0; t < CJn; t += 4) {
        float4 a = *reinterpret_cast<const float4*>(jf + t);
        float4 w = *reinterpret_cast<const float4*>(kw + t);
        acc += a.x * w.x + a.y * w.y + a.z * w.z + a.w * w.w;
    }
    kmat[g] = acc;
    sK[threadIdx.x] = acc;
    __syncthreads();

    const int base = threadIdx.x & ~15;
    const int x = e >> 2, z = e & 3;
    float k2 = 0.f;
#pragma unroll
    for (int y = 0; y < 4; ++y)
        k2 += sK[base + x * 4 + y] * sK[base + y * 4 + z];
    kmat2[g] = k2;
}

// ---------------------------------------------------------------------------
// Kernel 2: per joint j, GEMM  D[d, n] = sum_{k<768} Wcat[j][d][k] * X[k][n]
//   k in [0,256)   : X = Fp[c]
//   k in [256,512) : X = (K  . Fp[c])   generated on the fly from LDS K
//   k in [512,768) : X = (K^2 . Fp[c])  generated on the fly from LDS K^2
// Block tile M=128 x N=128; 8 waves (2x4), wave tile 64x32 -> 8 v8f accum.
// LDS tiles stored [row][APAD] with k contiguous so each v2f WMMA fragment
// is one aligned ds_load_b64 (no register re-pairing movs).
// W tile staged with GLOBAL_LOAD_ASYNC_TO_LDS_B128 when available.
// ---------------------------------------------------------------------------
__global__ __launch_bounds__(256)
void ft_gemm_kernel(const float* __restrict__ link, const float* __restrict__ W,
                    const int* __restrict__ parent_idx,
                    const int* __restrict__ child_idx,
                    const float* __restrict__ kmat, const float* __restrict__ kmat2,
                    float* __restrict__ out)
{
    __shared__ float A_lds[128 * APAD];   // A tile, [m][k]
    __shared__ float X_lds[128 * APAD];   // X tile, [n][k]
    __shared__ float sK[8 * 16];
    __shared__ float sK2[8 * 16];

    const int blk = blockIdx.x;          // 31*32 blocks
    const int j   = blk >> 5;
    const int rem = blk & 31;
    const int mt  = rem >> 4;            // 0..1  (M tile)
    const int nt  = rem & 15;            // 0..15 (N tile)
    const int m0  = mt * 128;
    const int b0  = nt * 8;              // 8 batches per block

    const int tid  = threadIdx.x;
    const int lane = tid & 31;
    const int wave = tid >> 5;
    const int wm   = wave >> 2;          // 0..1
    const int wn   = wave & 3;           // 0..3

    const int pl = parent_idx[j];        // parent link
    const int cl = child_idx[j];         // child link

    // Stage per-batch K and K^2 matrices
    {
        const int sel = tid >> 7;
        const int bb  = (tid >> 4) & 7;
        const int e   = tid & 15;
        const int p   = (b0 + bb) * Jn + j;
        if (sel) sK2[bb * 16 + e] = kmat2[p * 16 + e];
        else     sK [bb * 16 + e] = kmat [p * 16 + e];
    }
    __syncthreads();

    // Per-thread constant indices for X staging: n fixed, 8 contiguous k's
    const int xn  = tid & 127;           // n within tile
    const int xk0 = (tid >> 7) * 8;      // 0 or 8
    const int xbb = xn >> 4;             // batch within tile
    const int xs  = xn & 15;             // 4x4 element
    const int xr  = xs >> 2, zc = xs & 3;
    const float* LrowBase = link + (((size_t)(b0 + xbb) * Ln + pl) * Cn) * 16;

    // Fragment indices
    const int row   = lane & 15;
    const int khalf = (lane >> 4) * 2;   // A/B frags: lanes 0-15 K={0,1}, 16-31 K={2,3}

    v8f acc[4][2] = {};

    for (int kk = 0; kk < 768; kk += 16) {
        const int term = kk >> 8;
        const int c0   = kk & 255;

        if (kk + 16 < 768) {
            const int nterm = (kk + 16) >> 8;
            const float* w_pf = W + (((size_t)(j * 3 + nterm) * Cn) +
                                     (m0 + (tid >> 1))) * Cn + ((kk + 16) & 255);
            __builtin_prefetch(w_pf, 0, 0);
        }

        // ---- Stage A tile: W[j, term, m0+m, c0..c0+15] -> A_lds[m][0..15]
        const float* Wbase = W + ((size_t)(j * 3 + term) * Cn) * Cn;
#pragma unroll
        for (int q = tid; q < 512; q += 256) {      // 512 float4 quads
            const int m  = q >> 2;
            const int qq = q & 3;
            const float* gsrc = Wbase + (size_t)(m0 + m) * Cn + c0 + qq * 4;
            float* ldst = &A_lds[m * APAD + qq * 4];
#if HAVE_ASYNC_LDS
            async_copy16(gsrc, ldst);
#else
            *reinterpret_cast<float4*>(ldst) =
                *reinterpret_cast<const float4*>(gsrc);
#endif
        }

        // ---- Stage X tile: 8 contiguous k's at fixed n -> two b128 stores
        {
            const float* Lrow = LrowBase + (size_t)(c0 + xk0) * 16;
            float xv[8];
            if (term == 0) {
#pragma unroll
                for (int i = 0; i < 8; ++i) xv[i] = Lrow[i * 16 + xs];
            } else {
                const float* Km = (term == 1) ? (sK + xbb * 16) : (sK2 + xbb * 16);
                const float k0v = Km[xr * 4 + 0], k1v = Km[xr * 4 + 1];
                const float k2v = Km[xr * 4 + 2], k3v = Km[xr * 4 + 3];
#pragma unroll
                for (int i = 0; i < 8; ++i) {
                    const float* Lp = Lrow + i * 16;
                    xv[i] = k0v * Lp[zc] + k1v * Lp[4 + zc]
                          + k2v * Lp[8 + zc] + k3v * Lp[12 + zc];
                }
            }
            *reinterpret_cast<float4*>(&X_lds[xn * APAD + xk0]) =
                make_float4(xv[0], xv[1], xv[2], xv[3]);
            *reinterpret_cast<float4*>(&X_lds[xn * APAD + xk0 + 4]) =
                make_float4(xv[4], xv[5], xv[6], xv[7]);
        }
#if HAVE_ASYNC_LDS
        wait_async0();
#endif
        __syncthreads();

        // ---- Compute: 4 K-steps of V_WMMA_F32_16X16X4_F32
#pragma unroll
        for (int ks = 0; ks < 4; ++ks) {
            const int kb = ks * 4 + khalf;
            v2f afrag[4], bfrag[2];
#pragma unroll
            for (int mi = 0; mi < 4; ++mi)
                afrag[mi] = *reinterpret_cast<const v2f*>(
                    &A_lds[(wm * 64 + mi * 16 + row) * APAD + kb]);
#pragma unroll
            for (int ni = 0; ni < 2; ++ni)
                bfrag[ni] = *reinterpret_cast<const v2f*>(
                    &X_lds[(wn * 32 + ni * 16 + row) * APAD + kb]);
#pragma unroll
            for (int mi = 0; mi < 4; ++mi)
#pragma unroll
                for (int ni = 0; ni < 2; ++ni)
                    acc[mi][ni] = __builtin_amdgcn_wmma_f32_16x16x4_f32(
                        false, afrag[mi], false, bfrag[ni],
                        (short)0, acc[mi][ni], false, false);
        }
        __syncthreads();
    }

    // ---- Epilogue: comb = F_trans + Fc, raw to out[:, child]
    // C/D layout: VGPR r -> M = r + 8*(lane/16), N = lane%16
    const int ncol = lane & 15;
    const int mh   = (lane >> 4) * 8;
#pragma unroll
    for (int mi = 0; mi < 4; ++mi) {
#pragma unroll
        for (int ni = 0; ni < 2; ++ni) {
            const int n_g = nt * 128 + wn * 32 + ni * 16 + ncol;
            const int bb  = n_g >> 4;
            const int s   = n_g & 15;
#pragma unroll
            for (int r = 0; r < 8; ++r) {
                const int d = m0 + wm * 64 + mi * 16 + r + mh;
                const size_t idx = (((size_t)bb * Ln + cl) * Cn + d) * 16 + s;
                out[idx] = acc[mi][ni][r] + link[idx];
            }
        }
    }
}

// ---------------------------------------------------------------------------
// Kernel 3: LayerNorm over 4096 features per (b, link), in place on d_out.
// ---------------------------------------------------------------------------
__global__ __launch_bounds__(256)
void ln_kernel(const float* __restrict__ link, const float* __restrict__ gamma,
               const float* __restrict__ beta, float* __restrict__ out)
{
    __shared__ float red[16];
    __shared__ float stats[2];
    const int l = blockIdx.x & 31;
    const int b = blockIdx.x >> 5;
    const int tid = threadIdx.x;

    const float* src = (l == 0)
        ? (link + ((size_t)b * Ln + 0) * FEAT)
        : (out  + ((size_t)b * Ln + l) * FEAT);

    float4 v[4];
    float s1 = 0.f, s2 = 0.f;
#pragma unroll
    for (int i = 0; i < 4; ++i) {
        v[i] = reinterpret_cast<const float4*>(src)[tid * 4 + i];
        s1 += v[i].x + v[i].y + v[i].z + v[i].w;
        s2 += v[i].x * v[i].x + v[i].y * v[i].y + v[i].z * v[i].z + v[i].w * v[i].w;
    }
#pragma unroll
    for (int off = 16; off >= 1; off >>= 1) {
        s1 += __shfl_xor(s1, off, 32);
        s2 += __shfl_xor(s2, off, 32);
    }
    if ((tid & 31) == 0) {
        red[(tid >> 5) * 2 + 0] = s1;
        red[(tid >> 5) * 2 + 1] = s2;
    }
    __syncthreads();
    if (tid == 0) {
        float t1 = 0.f, t2 = 0.f;
#pragma unroll
        for (int w = 0; w < 8; ++w) { t1 += red[w * 2]; t2 += red[w * 2 + 1]; }
        const float mean = t1 * (1.0f / FEAT);
        const float var  = t2 * (1.0f / FEAT) - mean * mean;
        stats[0] = mean;
        stats[1] = rsqrtf(var + 1e-5f);
    }
    __syncthreads();
    const float mean = stats[0], inv = stats[1];

    float* dst = out + ((size_t)b * Ln + l) * FEAT;
    const float* g  = gamma + (size_t)l * FEAT;
    const float* bt = beta  + (size_t)l * FEAT;
#pragma unroll
    for (int i = 0; i < 4; ++i) {
        const float4 gv = reinterpret_cast<const float4*>(g)[tid * 4 + i];
        const float4 bv = reinterpret_cast<const float4*>(bt)[tid * 4 + i];
        float4 o;
        o.x = (v[i].x - mean) * inv * gv.x + bv.x;
        o.y = (v[i].y - mean) * inv * gv.y + bv.y;
        o.z = (v[i].z - mean) * inv * gv.z + bv.z;
        o.w = (v[i].w - mean) * inv * gv.w + bv.w;
        reinterpret_cast<float4*>(dst)[tid * 4 + i] = o;
    }
}

// ---------------------------------------------------------------------------
extern "C" void kernel_launch(void* const* d_in, const int* in_sizes, int n_in,
                              void* d_out, int out_size, void* d_ws, size_t ws_size,
                              hipStream_t stream) {
    const float* link   = (const float*)d_in[0];   // (B, L, C, 4, 4)
    const float* joint  = (const float*)d_in[1];   // (B, J, CJ)
    const float* W      = (const float*)d_in[2];   // (J, 3, C, C)
    const float* Kw     = (const float*)d_in[3];   // (J, 16, CJ)
    const float* Kb     = (const float*)d_in[4];   // (J, 16)
    const float* gamma  = (const float*)d_in[5];   // (L, 4096)
    const float* beta   = (const float*)d_in[6];   // (L, 4096)
    const int*   parent = (const int*)d_in[7];     // (J,)
    const int*   child  = (const int*)d_in[8];     // (J,)
    float* out = (float*)d_out;

    float* kmat  = (float*)d_ws;                   // B*J*16 floats
    float* kmat2 = kmat + (size_t)Bn * Jn * 16;    // B*J*16 floats

    kmat_kernel<<<(Bn * Jn * 16) / 256, 256, 0, stream>>>(joint, Kw, Kb, kmat, kmat2);
    ft_gemm_kernel<<<Jn * 32, 256, 0, stream>>>(link, W, parent, child, kmat, kmat2, out);
    ln_kernel<<<Bn * Ln, 256, 0, stream>>>(link, gamma, beta, out);
}